// CapsuleLayer_6184752906551
// MI455X (gfx1250) — compile-verified
//
#include <hip/hip_runtime.h>

// CapsuleLayer dynamic routing, fused for MI455X (gfx1250, wave32).
//
// - u_hat (537 MB) is never materialized: recomputed per routing pass with
//   V_WMMA_F32_16X16X4_F32 (exact fp32, K=8 as two K=4 steps).
// - routing logits reconstructed as u_hat.(v0+v1): only 256KB "V" + 256KB "s".
// - d-reduction for logits via DPP row_ror rotation adds (VALU, no LDS perm).
// - softmax held in registers; only per-m row-sums cross waves, via ds_add_f32
//   atomics into a parity-double-buffered Rsum[2][16] (2 barriers per capsule).
// - pass 0 (V==0) is exactly uniform c=1/32: pure GEMM fast path, no barriers.

typedef float v2f __attribute__((ext_vector_type(2)));
typedef float v8f __attribute__((ext_vector_type(8)));

#define BATCH   128
#define INCAPS  2048
#define OUTCAPS 32
#define INDIM   8
#define OUTDIM  16
#define NCHUNK  16            // capsules per block
#define BT      16            // batches per block (one WMMA M-tile)
#define SSIZE   (BATCH * OUTCAPS * OUTDIM)   // 65536 floats

// 16-lane (DPP row) rotation-add: after ror 8,4,2,1 every lane has the row sum.
template<int CTRL>
__device__ __forceinline__ float rotadd(float v) {
  int r = __builtin_amdgcn_update_dpp(0, __builtin_bit_cast(int, v),
                                      CTRL, 0xf, 0xf, true);
  return v + __builtin_bit_cast(float, r);
}
__device__ __forceinline__ float rowsum16(float v) {
  v = rotadd<0x128>(v);   // row_ror:8
  v = rotadd<0x124>(v);   // row_ror:4
  v = rotadd<0x122>(v);   // row_ror:2
  v = rotadd<0x121>(v);   // row_ror:1
  return v;
}

// ---------------------------------------------------------------- zero s & V
__global__ __launch_bounds__(256) void caps_zero(float* ws) {
  int i = blockIdx.x * 256 + threadIdx.x;     // grid 256 -> 65536 threads
  ws[i] = 0.0f;                                // s
  ws[i + SSIZE] = 0.0f;                        // V (sum of v's)
}

// ---------------------------------------------------------------- routing pass
// grid: (INCAPS/NCHUNK, BATCH/BT), block: 256 (8 waves; wave w owns o = 4w..4w+3)
__global__ __launch_bounds__(256) void caps_pass(const float* __restrict__ x,
                                                 const float* __restrict__ W,
                                                 const float* __restrict__ Vg,
                                                 float* __restrict__ s,
                                                 int uniform, float scale) {
  const int n0   = blockIdx.x * NCHUNK;
  const int bt   = blockIdx.y;
  const int tid  = threadIdx.x;
  const int wave = tid >> 5;
  const int lane = tid & 31;
  const int half = lane >> 4;      // 0: lanes 0-15, 1: lanes 16-31
  const int lh   = lane & 15;      // d-index (N of the WMMA tile)

  __shared__ float Vl[BT * OUTCAPS * OUTDIM];     // V tile [m][o][d], 32 KB
  __shared__ float Xl[BT][NCHUNK * INDIM + 2];    // x tile, padded vs bank conflicts
  __shared__ float Rsum[2][BT];                   // softmax row sums, parity-buffered

  if (!uniform) {
    for (int j = tid; j < BT * OUTCAPS * OUTDIM; j += 256)
      Vl[j] = Vg[(size_t)bt * (BT * OUTCAPS * OUTDIM) + j];
  }
  for (int j = tid; j < BT * NCHUNK * INDIM; j += 256) {
    int m = j >> 7;                 // NCHUNK*INDIM == 128
    int q = j & 127;
    Xl[m][q] = x[(size_t)(bt * BT + m) * (INCAPS * INDIM) + (size_t)n0 * INDIM + q];
  }
  if (tid < BT) Rsum[0][tid] = 0.0f;
  __syncthreads();

  const int o0 = wave * 4;
  v8f sacc[4];
  #pragma unroll
  for (int oi = 0; oi < 4; ++oi) sacc[oi] = (v8f){0,0,0,0,0,0,0,0};

  for (int nn = 0; nn < NCHUNK; ++nn) {
    const int n  = n0 + nn;
    const int p  = nn & 1;
    const int kb = half * 2;        // f32 frags: lanes<16 hold K={0,1}, lanes>=16 K={2,3}

    v2f A0, A1;                     // A 16x4 layout: M=lh, VGPR0=K(kb), VGPR1=K(kb+1)
    A0.x = Xl[lh][nn * INDIM + kb];
    A0.y = Xl[lh][nn * INDIM + kb + 1];
    A1.x = Xl[lh][nn * INDIM + 4 + kb];
    A1.y = Xl[lh][nn * INDIM + 4 + kb + 1];

    v8f C[4];
    float E[4][8];
    #pragma unroll
    for (int oi = 0; oi < 4; ++oi) {
      const int o = o0 + oi;
      const float* Wp = W + ((size_t)n * OUTCAPS + o) * (INDIM * OUTDIM);  // [k][d]
      v2f B0, B1;                   // B 4x16 layout: N=lh, rows split across halves
      B0.x = Wp[(kb    ) * OUTDIM + lh];
      B0.y = Wp[(kb + 1) * OUTDIM + lh];
      B1.x = Wp[(kb + 4) * OUTDIM + lh];
      B1.y = Wp[(kb + 5) * OUTDIM + lh];

      v8f c = (v8f){0,0,0,0,0,0,0,0};
      c = __builtin_amdgcn_wmma_f32_16x16x4_f32(false, A0, false, B0, (short)0, c, false, false);
      c = __builtin_amdgcn_wmma_f32_16x16x4_f32(false, A1, false, B1, (short)0, c, false, false);
      C[oi] = c;                    // u_hat tile: reg e = row m=(half*8+e), lane lh = d

      if (!uniform) {
        // logit[m][o] = sum_d u_hat[m,o,d]*V[m,o,d]; DPP reduce over the 16 d-lanes.
        // After rowsum16 every lane holds the logit; exp it in-register.
        #pragma unroll
        for (int e = 0; e < 8; ++e) {
          float r = c[e] * Vl[(half * 8 + e) * (OUTCAPS * OUTDIM) + o * OUTDIM + lh];
          E[oi][e] = __expf(rowsum16(r));
        }
      }
    }

    if (uniform) {
      // pass 0: c == 1/32 exactly (folded into flush scale); no softmax needed
      #pragma unroll
      for (int oi = 0; oi < 4; ++oi)
        #pragma unroll
        for (int e = 0; e < 8; ++e)
          sacc[oi][e] += C[oi][e];
      continue;
    }

    // per-m partial row-sums over this wave's 4 o's; cross-wave via LDS atomics
    float P[8];
    #pragma unroll
    for (int e = 0; e < 8; ++e)
      P[e] = (E[0][e] + E[1][e]) + (E[2][e] + E[3][e]);
    if (lh == 0) {                  // lane 0 -> m=0..7, lane 16 -> m=8..15
      const int mb = half * 8;
      atomicAdd(&Rsum[p][mb + 0], P[0]); atomicAdd(&Rsum[p][mb + 1], P[1]);
      atomicAdd(&Rsum[p][mb + 2], P[2]); atomicAdd(&Rsum[p][mb + 3], P[3]);
      atomicAdd(&Rsum[p][mb + 4], P[4]); atomicAdd(&Rsum[p][mb + 5], P[5]);
      atomicAdd(&Rsum[p][mb + 6], P[6]); atomicAdd(&Rsum[p][mb + 7], P[7]);
    }
    __syncthreads();

    // apply: sacc[m,o,d] += (E/Rsum[m]) * u_hat[m,o,d]
    #pragma unroll
    for (int e = 0; e < 8; ++e) {
      float rinv = __builtin_amdgcn_rcpf(Rsum[p][half * 8 + e]);
      #pragma unroll
      for (int oi = 0; oi < 4; ++oi)
        sacc[oi][e] += (E[oi][e] * rinv) * C[oi][e];
    }
    if (tid < BT) Rsum[1 - p][tid] = 0.0f;   // prep other parity for next capsule
    __syncthreads();
  }

  // flush partial s (reduced over this block's 16 capsules) to global
  #pragma unroll
  for (int oi = 0; oi < 4; ++oi) {
    const int o = o0 + oi;
    #pragma unroll
    for (int e = 0; e < 8; ++e) {
      const int m = half * 8 + e;
      atomicAdd(&s[(size_t)(bt * BT + m) * (OUTCAPS * OUTDIM) + o * OUTDIM + lh],
                sacc[oi][e] * scale);
    }
  }
}

// ---------------------------------------------------------------- squash + V update
// grid: BATCH blocks, 512 threads: t -> (o = t>>4, d = t&15); each 16-lane group = one o
__global__ __launch_bounds__(512) void caps_squash(float* __restrict__ s,
                                                   float* __restrict__ Vg,
                                                   float* __restrict__ out,
                                                   int last) {
  const int b = blockIdx.x;
  const int t = threadIdx.x;
  float val = s[b * 512 + t];
  float sq  = rowsum16(val * val);              // sum over the 16 d-lanes (DPP)
  float sc  = (sq / (1.0f + sq)) * rsqrtf(sq + 1e-8f);
  float v = sc * val;
  if (last) {
    out[b * 512 + t] = v;
  } else {
    Vg[b * 512 + t] += v;           // V = v0 (+ v1): next pass's logit weights
    s[b * 512 + t] = 0.0f;          // re-zero s for the next routing pass
  }
}

// ---------------------------------------------------------------- launch
extern "C" void kernel_launch(void* const* d_in, const int* in_sizes, int n_in,
                              void* d_out, int out_size, void* d_ws, size_t ws_size,
                              hipStream_t stream) {
  const float* x = (const float*)d_in[0];   // [128][2048][8]
  const float* W = (const float*)d_in[1];   // [2048][32][8][16]
  float* s   = (float*)d_ws;                // [128][32][16]
  float* Vg  = s + SSIZE;                   // [128][32][16]
  float* out = (float*)d_out;               // [128][32][16]

  caps_zero<<<256, 256, 0, stream>>>(s);    // zero s and V
  for (int it = 0; it < 3; ++it) {
    caps_pass<<<dim3(INCAPS / NCHUNK, BATCH / BT), 256, 0, stream>>>(
        x, W, Vg, s, it == 0 ? 1 : 0, it == 0 ? (1.0f / OUTCAPS) : 1.0f);
    caps_squash<<<BATCH, 512, 0, stream>>>(s, Vg, out, it == 2);
  }
}